// UniDirAttention_4217657885104
// MI455X (gfx1250) — compile-verified
//
#include <hip/hip_runtime.h>
#include <hip/hip_bf16.h>
#include <stdint.h>

typedef __attribute__((ext_vector_type(2))) float v2f;
typedef __attribute__((ext_vector_type(8))) float v8f;

#define B_      32
#define L_      4096
#define D_      1024
#define NCHUNK  32
#define LC      (L_ / NCHUNK)      // 128 rows per block
#define TROWS   16                 // rows per tile (WMMA M)
#define NTILES  (LC / TROWS)       // 8 tiles per block
#define TSTR    (D_ + 4)           // padded LDS row stride (floats) -> conflict-free row access
#define NEG_INF_F (-1e30f)

#define SM_BUF    (TROWS * TSTR)                           // 16448 floats per tile buffer
#define SM_FLOATS (2 * SM_BUF + D_ + 8 * 16 + 16 + 16 + 4) // 34084 floats
#define SM_BYTES  (SM_FLOATS * 4)                          // ~136 KB dynamic LDS

// Async-copy one 16x1024 fp32 tile (row-major, padded stride in LDS).
// Each thread moves 16B per row; 16 async ops per wave per tile (ASYNCcnt += 16).
__device__ __forceinline__ void async_tile_load(float* buf, const float* gsrc_row0, int tid) {
#pragma unroll
  for (int r = 0; r < TROWS; ++r) {
    unsigned lds = (unsigned)(size_t)(&buf[r * TSTR + tid * 4]);
    uint64_t ga  = (uint64_t)(size_t)(gsrc_row0 + (size_t)r * D_ + tid * 4);
    asm volatile("global_load_async_to_lds_b128 %0, %1, off" :: "v"(lds), "v"(ga) : "memory");
  }
}

__global__ void __launch_bounds__(256) attn_pass1(
    const float* __restrict__ seq, const float* __restrict__ vecg,
    const int* __restrict__ mask, float* __restrict__ scores_out,
    float* __restrict__ mzWS, float* __restrict__ accWS) {
  extern __shared__ float sm[];
  float* buf0    = sm;
  float* buf1    = sm + SM_BUF;
  float* vls     = sm + 2 * SM_BUF;      // vector[b,:] (1024 floats)
  float* partial = vls + D_;             // [8 waves][16 rows] partial scores
  float* sarr    = partial + 8 * 16;     // 16 masked scores of current tile
  float* wexp    = sarr + 16;            // 16 exp weights of current tile
  float* stats   = wexp + 16;            // [0]=running m, [1]=running Z, [2]=rescale

  const int tid  = threadIdx.x;
  const int lane = tid & 31;
  const int wv   = tid >> 5;
  const int b     = blockIdx.x >> 5;         // NCHUNK == 32
  const int chunk = blockIdx.x & (NCHUNK - 1);
  const int row0  = chunk * LC;
  const float* seqB = seq + (size_t)b * L_ * D_;

  if (tid == 0) { stats[0] = NEG_INF_F; stats[1] = 0.0f; stats[2] = 0.0f; }

  const int col = tid * 4;                   // this thread's 4 pooled columns
  float acc0 = 0.f, acc1 = 0.f, acc2 = 0.f, acc3 = 0.f;

  // Prologue: async-load vector[b,:] and tile 0, then drain ASYNCcnt.
  {
    unsigned lds = (unsigned)(size_t)(&vls[col]);
    uint64_t ga  = (uint64_t)(size_t)(vecg + (size_t)b * D_ + col);
    asm volatile("global_load_async_to_lds_b128 %0, %1, off" :: "v"(lds), "v"(ga) : "memory");
  }
  async_tile_load(buf0, seqB + (size_t)row0 * D_, tid);
  asm volatile("s_wait_asynccnt 0x0" ::: "memory");
  __syncthreads();

  // WMMA A/B lane layout for 16x16x4 f32: lanes 0-15 hold K=0,1; lanes 16-31 hold K=2,3.
  const int m16  = lane & 15;
  const int koff = (lane >> 4) << 1;

  for (int t = 0; t < NTILES; ++t) {
    float* cur = (t & 1) ? buf1 : buf0;
    if (t + 1 < NTILES) {
      float* nxt = ((t + 1) & 1) ? buf1 : buf0;
      async_tile_load(nxt, seqB + (size_t)(row0 + (t + 1) * TROWS) * D_, tid);
      // allow next tile's 16 async ops to remain in flight; guarantee current tile complete
      asm volatile("s_wait_asynccnt 0x10" ::: "memory");
    } else {
      asm volatile("s_wait_asynccnt 0x0" ::: "memory");
    }
    __syncthreads();

    // --- scores[row] = <seq[row,:], vec> via V_WMMA_F32_16X16X4_F32 ---
    // Each wave covers a 128-wide K slice; B columns all equal the vector chunk,
    // so C column 0 carries the partial scores for the 16 rows.
    const float* arow  = cur + m16 * TSTR + koff;
    const float* bcol  = vls + koff;
    const int    kbase = wv * (D_ / 8);
    v8f c = {0.f, 0.f, 0.f, 0.f, 0.f, 0.f, 0.f, 0.f};
#pragma unroll
    for (int kk = 0; kk < D_ / 8; kk += 4) {
      const int k = kbase + kk;
      v2f a;  a.x  = arow[k]; a.y  = arow[k + 1];
      v2f bb; bb.x = bcol[k]; bb.y = bcol[k + 1];
      c = __builtin_amdgcn_wmma_f32_16x16x4_f32(false, a, false, bb, (short)0, c,
                                                false, false);
    }
    // C layout: lane 0 holds rows 0..7 of column 0 in VGPRs 0..7; lane 16 rows 8..15.
    if (lane == 0) {
#pragma unroll
      for (int r = 0; r < 8; ++r) partial[wv * 16 + r] = c[r];
    } else if (lane == 16) {
#pragma unroll
      for (int r = 0; r < 8; ++r) partial[wv * 16 + 8 + r] = c[r];
    }
    __syncthreads();

    // Cross-wave reduce + mask; store masked score for the final weights pass.
    if (tid < 16) {
      float s = 0.f;
#pragma unroll
      for (int w = 0; w < 8; ++w) s += partial[w * 16 + tid];
      const int rowg = row0 + t * TROWS + tid;
      s = (mask[(size_t)b * L_ + rowg] > 0) ? s : NEG_INF_F;
      scores_out[(size_t)b * L_ + rowg] = s;
      sarr[tid] = s;
    }
    __syncthreads();

    // Online softmax bookkeeping (tiny serial work).
    if (tid == 0) {
      const float mOld = stats[0];
      float tmax = sarr[0];
#pragma unroll
      for (int r = 1; r < 16; ++r) tmax = fmaxf(tmax, sarr[r]);
      const float mNew = fmaxf(mOld, tmax);
      const float sc   = __expf(mOld - mNew);
      float zs = 0.f;
#pragma unroll
      for (int r = 0; r < 16; ++r) {
        const float e = __expf(sarr[r] - mNew);
        wexp[r] = e;
        zs += e;
      }
      stats[0] = mNew;
      stats[1] = stats[1] * sc + zs;
      stats[2] = sc;
    }
    __syncthreads();

    // Rescale + accumulate weighted rows from the same LDS tile (single HBM pass).
    const float sc = stats[2];
    acc0 *= sc; acc1 *= sc; acc2 *= sc; acc3 *= sc;
#pragma unroll
    for (int r = 0; r < TROWS; ++r) {
      const float w = wexp[r];
      const float* rp = cur + r * TSTR + col;
      acc0 = fmaf(w, rp[0], acc0);
      acc1 = fmaf(w, rp[1], acc1);
      acc2 = fmaf(w, rp[2], acc2);
      acc3 = fmaf(w, rp[3], acc3);
    }
    __syncthreads();  // tile buffer may be overwritten by next iteration's async issue
  }

  if (tid == 0) {
    mzWS[((size_t)b * NCHUNK + chunk) * 2 + 0] = stats[0];
    mzWS[((size_t)b * NCHUNK + chunk) * 2 + 1] = stats[1];
  }
  float4* ap = (float4*)(accWS + ((size_t)b * NCHUNK + chunk) * D_ + col);
  *ap = make_float4(acc0, acc1, acc2, acc3);
}

__global__ void __launch_bounds__(256) attn_pass2(
    const float* __restrict__ mzWS, const float* __restrict__ accWS,
    float* __restrict__ out) {
  const int b   = blockIdx.x;
  const int tid = threadIdx.x;

  // Log-sum-exp merge of the 32 chunk partials (redundant per-thread, trivial cost).
  float mb = NEG_INF_F;
#pragma unroll
  for (int c = 0; c < NCHUNK; ++c)
    mb = fmaxf(mb, mzWS[((size_t)b * NCHUNK + c) * 2]);
  float Zb = 0.f;
#pragma unroll
  for (int c = 0; c < NCHUNK; ++c)
    Zb += mzWS[((size_t)b * NCHUNK + c) * 2 + 1] *
          __expf(mzWS[((size_t)b * NCHUNK + c) * 2] - mb);
  const float invZ = 1.0f / Zb;

  // pooled[b, col..col+3]
  const int col = tid * 4;
  float p0 = 0.f, p1 = 0.f, p2 = 0.f, p3 = 0.f;
  for (int c = 0; c < NCHUNK; ++c) {
    const float e = __expf(mzWS[((size_t)b * NCHUNK + c) * 2] - mb);
    const float4 a = *(const float4*)(accWS + ((size_t)b * NCHUNK + c) * D_ + col);
    p0 = fmaf(a.x, e, p0);
    p1 = fmaf(a.y, e, p1);
    p2 = fmaf(a.z, e, p2);
    p3 = fmaf(a.w, e, p3);
  }
  float* pout = out + (size_t)b * D_ + col;
  pout[0] = p0 * invZ; pout[1] = p1 * invZ;
  pout[2] = p2 * invZ; pout[3] = p3 * invZ;

  // weights: transform masked scores (written by pass 1) in place.
  float* wout = out + (size_t)B_ * D_ + (size_t)b * L_;
  for (int l = tid; l < L_; l += 256)
    wout[l] = __expf(wout[l] - mb) * invZ;
}

extern "C" void kernel_launch(void* const* d_in, const int* in_sizes, int n_in,
                              void* d_out, int out_size, void* d_ws, size_t ws_size,
                              hipStream_t stream) {
  const float* seq  = (const float*)d_in[0];
  const float* vec  = (const float*)d_in[1];
  const int*   mask = (const int*)d_in[2];
  float* out = (float*)d_out;

  // Workspace layout: [m,Z pairs: B*NCHUNK*2 floats][acc: B*NCHUNK*D floats]
  float* mz  = (float*)d_ws;
  float* acc = mz + 2 * B_ * NCHUNK;

  hipFuncSetAttribute(reinterpret_cast<const void*>(attn_pass1),
                      hipFuncAttributeMaxDynamicSharedMemorySize, SM_BYTES);

  attn_pass1<<<dim3(B_ * NCHUNK), dim3(256), SM_BYTES, stream>>>(
      seq, vec, mask, out + (size_t)B_ * D_, mz, acc);
  attn_pass2<<<dim3(B_), dim3(256), 0, stream>>>(mz, acc, out);
}